// RoutingCapsule_70841190580940
// MI455X (gfx1250) — compile-verified
//
#include <hip/hip_runtime.h>
#include <math.h>

// Problem constants (from reference setup_inputs)
#define B_TOT   32
#define NIN     2048
#define DIN     16
#define NOUT    64
#define DOUT    32
#define JD      (NOUT * DOUT)   // 2048 flattened (j,d) rows
#define BT      8               // batch tile: u_hat slab = 2048*2048*8*4B = 128MB < 192MB L2
#define NTILES  (B_TOT / BT)
#define ITERS   3

typedef float v2f __attribute__((ext_vector_type(2)));
typedef float v8f __attribute__((ext_vector_type(8)));

// ---------------------------------------------------------------------------
// K1: u_hat[i][r][br] = sum_k W[i][r][k] * x[b0+br][i][k]      (r = j*32+d)
// fp32 WMMA 16x16x4, K=16 chained as 4 steps. One wave = 16 rows x 16 cols
// (8 valid batch cols). Block = 8 waves = 128 rows; 16 blocks per i.
// W is streamed non-temporally so the L2 stays owned by the u_hat slab.
// ---------------------------------------------------------------------------
__global__ __launch_bounds__(256) void KUhat(const float* __restrict__ W,
                                             const float* __restrict__ X,
                                             float* __restrict__ UH, int b0) {
  const int lane = threadIdx.x & 31;
  const int wave = threadIdx.x >> 5;
  const int i    = blockIdx.x >> 4;                       // input capsule
  const int r0   = ((blockIdx.x & 15) << 7) + (wave << 4); // row-tile base
  const int half = lane >> 4;   // 0: K pair {0,1}; 1: K pair {2,3}
  const int lrow = lane & 15;   // A: row M; B: col N; D: col N

  // A: lane holds W[i][r0+lrow][kb + 2*half .. +1]  (contiguous 8B)
  const float* Wrow = W + ((size_t)i * JD + (r0 + lrow)) * DIN;
  // B[k][n] = x[b0+n][i][k]; clamp n>=BT (those columns are discarded)
  const int nclamp = lrow < BT ? lrow : (BT - 1);
  const float* Xcol = X + ((size_t)(b0 + nclamp) * NIN + i) * DIN;

  v8f acc = {0.f, 0.f, 0.f, 0.f, 0.f, 0.f, 0.f, 0.f};
#pragma unroll
  for (int kb = 0; kb < DIN; kb += 4) {
    v2f a, b;
    a.x = __builtin_nontemporal_load(Wrow + kb + 2 * half + 0);
    a.y = __builtin_nontemporal_load(Wrow + kb + 2 * half + 1);
    b.x = Xcol[kb + 2 * half + 0];
    b.y = Xcol[kb + 2 * half + 1];
    // D = A(16x4 f32) * B(4x16 f32) + C   -> v_wmma_f32_16x16x4_f32
    acc = __builtin_amdgcn_wmma_f32_16x16x4_f32(
        /*neg_a=*/false, a, /*neg_b=*/false, b,
        /*c_mod=*/(short)0, acc, /*reuse_a=*/false, /*reuse_b=*/false);
  }

  // D layout: VGPR r -> lanes 0-15: (M=r, N=lane); lanes 16-31: (M=r+8, N=lane-16)
  if (lrow < BT) {
    float* dst = UH + ((size_t)i * JD + r0 + 8 * half) * BT + lrow;
#pragma unroll
    for (int r = 0; r < 8; ++r)
      dst[(size_t)r * BT] = acc[r];
  }
}

// ---------------------------------------------------------------------------
// K2: c[i][j][br] = softmax_j(b_log[i][j][br])   (uniform 1/64 on iteration 0)
// One thread per (i,br); 64 j-values live in unrolled registers.
// ---------------------------------------------------------------------------
__global__ __launch_bounds__(256) void KC(const float* __restrict__ BL,
                                          float* __restrict__ C, int first) {
  const int t  = blockIdx.x * blockDim.x + threadIdx.x;  // 16384 = NIN*BT
  const int i  = t >> 3;
  const int br = t & 7;
  float* cp = C + (size_t)i * (NOUT * BT) + br;
  if (first) {
#pragma unroll
    for (int j = 0; j < NOUT; ++j) cp[j * BT] = 1.0f / NOUT;
    return;
  }
  const float* blp = BL + (size_t)i * (NOUT * BT) + br;
  float vals[NOUT];
#pragma unroll
  for (int j = 0; j < NOUT; ++j) vals[j] = blp[j * BT];
  float m = vals[0];
#pragma unroll
  for (int j = 1; j < NOUT; ++j) m = fmaxf(m, vals[j]);
  float sum = 0.f;
#pragma unroll
  for (int j = 0; j < NOUT; ++j) { vals[j] = __expf(vals[j] - m); sum += vals[j]; }
  const float inv = 1.f / sum;
#pragma unroll
  for (int j = 0; j < NOUT; ++j) cp[j * BT] = vals[j] * inv;
}

// K3a: seed s[jd][br] with bias[jd] (s is accumulated by atomics in K3b)
__global__ __launch_bounds__(1024) void KInitS(float* __restrict__ S,
                                               const float* __restrict__ bias) {
  const int t = blockIdx.x * blockDim.x + threadIdx.x;   // 16384
  S[t] = bias[t >> 3];
}

// ---------------------------------------------------------------------------
// K3b: s[j][d][br] += sum_i c[i][j][br] * u_hat[i][jd][br]
// 16 element-blocks x 32 i-chunks (64 i each); u_hat loads fully coalesced
// and L2-resident. Partial sums folded with one atomicAdd per thread.
// ---------------------------------------------------------------------------
__global__ __launch_bounds__(1024) void KS(const float* __restrict__ C,
                                           const float* __restrict__ UH,
                                           float* __restrict__ S) {
  const int e  = (blockIdx.x & 15) * 1024 + threadIdx.x;  // 0..16383 = jd*8+br
  const int i0 = (blockIdx.x >> 4) * 64;                  // i-chunk base
  const int br = e & 7;
  const int j  = e >> 8;                                  // (e>>3)>>5
  const float* uh = UH + (size_t)i0 * (JD * BT) + e;      // +16384 per i
  const float* cp = C + (size_t)i0 * (NOUT * BT) + j * BT + br;  // +512 per i
  float sum = 0.f;
#pragma unroll 4
  for (int ii = 0; ii < 64; ++ii)
    sum += cp[(size_t)ii * (NOUT * BT)] * uh[(size_t)ii * (JD * BT)];
  atomicAdd(&S[e], sum);
}

// ---------------------------------------------------------------------------
// K4: v[b0+br][j][d] = squash(s[j][d][br]); written directly into d_out
// (d_out layout [B][Nout][Dout] == v layout; last iteration's write wins).
// ---------------------------------------------------------------------------
__global__ __launch_bounds__(256) void KSquash(const float* __restrict__ S,
                                               float* __restrict__ V, int b0) {
  const int t = blockIdx.x * blockDim.x + threadIdx.x;    // 512 = NOUT*BT
  if (t >= NOUT * BT) return;
  const int j = t >> 3, br = t & 7;
  const float* sp = S + (size_t)j * (DOUT * BT) + br;
  float nrm = 0.f;
#pragma unroll
  for (int d = 0; d < DOUT; ++d) { const float x = sp[d * BT]; nrm += x * x; }
  const float scale = (nrm / (1.f + nrm)) * rsqrtf(nrm + 1e-9f);
  float* vp = V + ((size_t)(b0 + br) * NOUT + j) * DOUT;
#pragma unroll
  for (int d = 0; d < DOUT; ++d) vp[d] = scale * sp[d * BT];
}

// ---------------------------------------------------------------------------
// K5: b_log[i][j][br] (+)= sum_d u_hat[i][j*32+d][br] * v[b0+br][j][d]
// One thread per (i,j,br); v row (128B) pulled as 8 float4s (cache-hot).
// Iteration 0 overwrites (b_log starts at 0 -> no init kernel needed).
// ---------------------------------------------------------------------------
__global__ __launch_bounds__(256) void KBlog(const float* __restrict__ UH,
                                             const float* __restrict__ V,
                                             float* __restrict__ BL,
                                             int b0, int first) {
  const int t  = blockIdx.x * blockDim.x + threadIdx.x;  // 0..NIN*NOUT*BT-1
  const int br = t & 7;
  const int j  = (t >> 3) & (NOUT - 1);
  const int i  = t >> 9;
  const float* uh = UH + ((size_t)i * JD + j * DOUT) * BT + br;
  const float4* vp = (const float4*)(V + ((size_t)(b0 + br) * NOUT + j) * DOUT);
  float acc = 0.f;
#pragma unroll
  for (int q = 0; q < 8; ++q) {
    const float4 v4 = vp[q];
    acc += uh[(q * 4 + 0) * BT] * v4.x + uh[(q * 4 + 1) * BT] * v4.y +
           uh[(q * 4 + 2) * BT] * v4.z + uh[(q * 4 + 3) * BT] * v4.w;
  }
  BL[t] = first ? acc : (BL[t] + acc);
}

// ---------------------------------------------------------------------------
extern "C" void kernel_launch(void* const* d_in, const int* in_sizes, int n_in,
                              void* d_out, int out_size, void* d_ws, size_t ws_size,
                              hipStream_t stream) {
  (void)in_sizes; (void)n_in; (void)out_size; (void)ws_size;
  const float* x    = (const float*)d_in[0];  // [32,2048,16]
  const float* W    = (const float*)d_in[1];  // [2048,64,32,16]
  const float* bias = (const float*)d_in[2];  // [64,32]
  float* out = (float*)d_out;                 // [32,64,32]

  char* ws = (char*)d_ws;
  const size_t UH_BYTES = (size_t)NIN * JD * BT * sizeof(float);   // 128 MB
  const size_t CB_BYTES = (size_t)NIN * NOUT * BT * sizeof(float); //   4 MB
  float* uhat = (float*)ws;
  float* cbuf = (float*)(ws + UH_BYTES);
  float* blog = (float*)(ws + UH_BYTES + CB_BYTES);
  float* sbuf = (float*)(ws + UH_BYTES + 2 * CB_BYTES);            //  64 KB

  for (int nt = 0; nt < NTILES; ++nt) {
    const int b0 = nt * BT;
    KUhat<<<NIN * 16, 256, 0, stream>>>(W, x, uhat, b0);
    for (int it = 0; it < ITERS; ++it) {
      KC    <<<64,  256,  0, stream>>>(blog, cbuf, it == 0);
      KInitS<<<16,  1024, 0, stream>>>(sbuf, bias);
      KS    <<<512, 1024, 0, stream>>>(cbuf, uhat, sbuf);
      KSquash<<<2,  256,  0, stream>>>(sbuf, out, b0);
      if (it < ITERS - 1)
        KBlog<<<(NIN * NOUT * BT) / 256, 256, 0, stream>>>(uhat, out, blog, b0, it == 0);
    }
  }
}